// HGAT_6949257085552
// MI455X (gfx1250) — compile-verified
//
#include <hip/hip_runtime.h>

typedef __attribute__((ext_vector_type(2))) float v2f;
typedef __attribute__((ext_vector_type(8))) float v8f;

#define TB 256
#define BM 128
#define BN 32
#define BK 32
#define ALD 36  // padded f32 LDS row: 144B (16B-aligned chunks, spread banks)

__device__ __forceinline__ int imin(int a, int b) { return a < b ? a : b; }

__device__ __forceinline__ void atomic_max_f32(float* a, float v) {
  if (v >= 0.0f) atomicMax((int*)a, __float_as_int(v));
  else           atomicMin((unsigned int*)a, __float_as_uint(v));
}

// ---------------- WMMA GEMM (f32): C = act(A[M,K] @ B[K,N] (+bias)) ----------------
// 256 threads = 8 waves; each wave owns one 16-row tile and both 16-col tiles.
// Full-precision V_WMMA_F32_16X16X4_F32 (8 steps per BK=32 tile, 2 col tiles).
// A panel staged with GLOBAL_LOAD_ASYNC_TO_LDS_B128 (ASYNCcnt) into double-buffered
// LDS on the fast path (no M/K tail, K%4==0); clamped VGPR fallback otherwise.
// B panel (32x32) staged transposed via VGPRs with one b128 LDS store per thread.
__global__ __launch_bounds__(TB) void k_gemm_wmma(
    const float* __restrict__ A, const float* __restrict__ B,
    const float* __restrict__ bias, float* __restrict__ C,
    int M, int N, int K, float slope, int act)
{
  __shared__ __align__(16) float Af[2][BM][ALD];
  __shared__ __align__(16) float Bt[2][BN][ALD];
  const int tid  = threadIdx.x;
  const int lane = tid & 31;
  const int wave = tid >> 5;          // 0..7 -> row tile
  const int m0   = blockIdx.x * BM;
  const int n0   = blockIdx.y * BN;
  const int wm   = wave * 16;
  const int fr   = lane & 15;
  const int kA   = (lane >> 4) * 2;   // f32 frag: K pair per lane half

  // A staging: 128 rows x 32 k f32; thread t copies 16 f32 (4 x b128)
  const int arow = tid >> 1;
  const int akk  = (tid & 1) * 16;
  // B staging: thread t: col bn, 4 k values -> one float4 into transposed row
  const int bn   = tid >> 3;
  const int bkk  = (tid & 7) * 4;

  const int nkt = (K + BK - 1) / BK;
  const bool kvec = (K & 3) == 0;

  auto issueA = [&](int it) {
    const int k0 = it * BK;
    const int b  = it & 1;
    const int gm = m0 + arow;
    if (kvec && (m0 + BM <= M) && (k0 + BK <= K)) {
      const float* gp = A + (size_t)gm * K + k0 + akk;
      const unsigned la = (unsigned)(uintptr_t)&Af[b][arow][akk];
      #pragma unroll
      for (int j = 0; j < 4; ++j) {
        asm volatile("global_load_async_to_lds_b128 %0, %1, off"
                     :: "v"(la + 16u * j), "v"(gp + 4 * j) : "memory");
      }
    } else {
      const size_t rowb = (size_t)imin(gm, M - 1) * K;
      #pragma unroll
      for (int j = 0; j < 16; ++j) {
        const int gk = k0 + akk + j;
        const float v = A[rowb + imin(gk, K - 1)];
        Af[b][arow][akk + j] = (gm < M && gk < K) ? v : 0.0f;
      }
    }
  };

  auto issueB = [&](int it) {
    const int k0  = it * BK;
    const int b   = it & 1;
    const int gn  = n0 + bn;
    const int gnc = imin(gn, N - 1);
    float rr[4];
    #pragma unroll
    for (int j = 0; j < 4; ++j) {
      const int gk = k0 + bkk + j;
      const float v = B[(size_t)imin(gk, K - 1) * N + gnc];
      rr[j] = (gk < K && gn < N) ? v : 0.0f;
    }
    float4 q; q.x = rr[0]; q.y = rr[1]; q.z = rr[2]; q.w = rr[3];
    *(float4*)&Bt[b][bn][bkk] = q;  // ds_store_b128 (transposed)
  };

  v8f acc0 = {}, acc1 = {};

  issueA(0);
  issueB(0);
  for (int it = 0; it < nkt; ++it) {
    const int b = it & 1;
    asm volatile("s_wait_asynccnt 0x0" ::: "memory");  // my async copies for tile `it` done
    __syncthreads();                                   // tile `it` published (dscnt waited by compiler)

    if (it + 1 < nkt) { issueA(it + 1); issueB(it + 1); }  // stream next tile into other buffer

    #pragma unroll
    for (int s = 0; s < 8; ++s) {
      v2f av = *(const v2f*)&Af[b][wm + fr][4 * s + kA];
      v2f b0 = *(const v2f*)&Bt[b][fr][4 * s + kA];
      v2f b1 = *(const v2f*)&Bt[b][16 + fr][4 * s + kA];
      acc0 = __builtin_amdgcn_wmma_f32_16x16x4_f32(false, av, false, b0, (short)0, acc0, false, false);
      acc1 = __builtin_amdgcn_wmma_f32_16x16x4_f32(false, av, false, b1, (short)0, acc1, false, false);
    }
    __syncthreads();  // all waves done reading buffer b before it is overwritten next round
  }

  const int mb  = m0 + wm + (lane >> 4) * 8;
  const int gn0 = n0 + fr;
  const int gn1 = n0 + 16 + fr;
  #pragma unroll
  for (int r = 0; r < 8; ++r) {
    const int gm = mb + r;
    if (gm >= M) continue;
    if (gn0 < N) {
      float v = acc0[r];
      if (bias) v += bias[gn0];
      if (act)  v = v > 0.0f ? v : v * slope;
      C[(size_t)gm * N + gn0] = v;
    }
    if (gn1 < N) {
      float v = acc1[r];
      if (bias) v += bias[gn1];
      if (act)  v = v > 0.0f ? v : v * slope;
      C[(size_t)gm * N + gn1] = v;
    }
  }
}

// ---------------- small kernels ----------------
__global__ void k_fill(float* __restrict__ p, float v, long n) {
  long i = (long)blockIdx.x * blockDim.x + threadIdx.x;
  if (i < n) p[i] = v;
}

__global__ void k_init_bias2(float* __restrict__ out, const float* __restrict__ b1,
                             const float* __restrict__ b2, long N, int F) {
  long i = (long)blockIdx.x * blockDim.x + threadIdx.x;
  if (i >= N * (long)F) return;
  int j = (int)(i % F);
  float v = b1[j];
  if (b2) v += b2[j];
  out[i] = v;
}

// v[h*K+k] = sum_c W[k, h*C+c] * a[h*C+c]
__global__ void k_make_v(const float* __restrict__ W, const float* __restrict__ a,
                         float* __restrict__ v, int K, int C, int H) {
  int i = blockIdx.x * blockDim.x + threadIdx.x;
  if (i >= H * K) return;
  int h = i / K, k = i - h * K;
  const float* wr = W + (size_t)k * (H * C) + (size_t)h * C;
  const float* ar = a + (size_t)h * C;
  float acc = 0.0f;
  for (int c = 0; c < C; ++c) acc += wr[c] * ar[c];
  v[i] = acc;
}

// H=2 fused: s[n*2+h] = sum_k x[n,k] * v[h*K+k]; reads x row once
__global__ void k_scores2(const float* __restrict__ x, const float* __restrict__ v,
                          float* __restrict__ s, int N, int K) {
  int n = blockIdx.x * blockDim.x + threadIdx.x;
  if (n >= N) return;
  const float* xr = x + (size_t)n * K;
  const float* v0 = v;
  const float* v1 = v + K;
  float a0 = 0.0f, a1 = 0.0f;
  if ((K & 3) == 0) {
    for (int k = 0; k < K; k += 4) {
      const float4 xv = *(const float4*)(xr + k);
      const float4 u0 = *(const float4*)(v0 + k);
      const float4 u1 = *(const float4*)(v1 + k);
      a0 += xv.x * u0.x + xv.y * u0.y + xv.z * u0.z + xv.w * u0.w;
      a1 += xv.x * u1.x + xv.y * u1.y + xv.z * u1.z + xv.w * u1.w;
    }
  } else {
    for (int k = 0; k < K; ++k) { a0 += xr[k] * v0[k]; a1 += xr[k] * v1[k]; }
  }
  s[(size_t)n * 2 + 0] = a0;
  s[(size_t)n * 2 + 1] = a1;
}

// H=2 fused: s[n*2+h] = sum_c hs[n, h*C+c] * a[h*C+c]
__global__ void k_scores_h2(const float* __restrict__ hs, const float* __restrict__ a,
                            float* __restrict__ s, int N, int C) {
  int n = blockIdx.x * blockDim.x + threadIdx.x;
  if (n >= N) return;
  const float* xr = hs + (size_t)n * 2 * C;
  float a0 = 0.0f, a1 = 0.0f;
  for (int c = 0; c < C; c += 4) {
    const float4 x0 = *(const float4*)(xr + c);
    const float4 x1 = *(const float4*)(xr + C + c);
    const float4 u0 = *(const float4*)(a + c);
    const float4 u1 = *(const float4*)(a + C + c);
    a0 += x0.x * u0.x + x0.y * u0.y + x0.z * u0.z + x0.w * u0.w;
    a1 += x1.x * u1.x + x1.y * u1.y + x1.z * u1.z + x1.w * u1.w;
  }
  s[(size_t)n * 2 + 0] = a0;
  s[(size_t)n * 2 + 1] = a1;
}

__global__ void k_selfloop_stats(const int* __restrict__ dst, const float* __restrict__ se,
                                 float* __restrict__ sesum, float* __restrict__ cnt,
                                 int E, int H) {
  int e = blockIdx.x * blockDim.x + threadIdx.x;
  if (e >= E) return;
  int d = dst[e];
  for (int h = 0; h < H; ++h) atomicAdd(&sesum[(size_t)d * H + h], se[(size_t)e * H + h]);
  atomicAdd(&cnt[d], 1.0f);
}

__global__ void k_selfloop_fin(const float* __restrict__ sesum, const float* __restrict__ cnt,
                               float* __restrict__ sslf, int N, int H) {
  int i = blockIdx.x * blockDim.x + threadIdx.x;
  if (i >= N * H) return;
  int n = i / H;
  sslf[i] = sesum[i] / fmaxf(cnt[n], 1.0f);
}

__global__ void k_logits_max(const int* __restrict__ src, const int* __restrict__ dst,
                             const float* __restrict__ ssrc, const float* __restrict__ sdst,
                             const float* __restrict__ se, float* __restrict__ al,
                             float* __restrict__ m, int E, int H) {
  int e = blockIdx.x * blockDim.x + threadIdx.x;
  if (e >= E) return;
  int s = src[e], d = dst[e];
  for (int h = 0; h < H; ++h) {
    float t = ssrc[(size_t)s * H + h] + sdst[(size_t)d * H + h] + se[(size_t)e * H + h];
    t = t > 0.0f ? t : 0.2f * t;
    al[(size_t)e * H + h] = t;
    atomic_max_f32(&m[(size_t)d * H + h], t);
  }
}

__global__ void k_logits_max_self(const float* __restrict__ ssrc, const float* __restrict__ sdst,
                                  const float* __restrict__ sslf, float* __restrict__ alslf,
                                  float* __restrict__ m, int N, int H) {
  int i = blockIdx.x * blockDim.x + threadIdx.x;
  if (i >= N * H) return;
  float t = ssrc[i] + sdst[i] + sslf[i];
  t = t > 0.0f ? t : 0.2f * t;
  alslf[i] = t;
  m[i] = fmaxf(m[i], t);  // prior edge kernel finished (stream order); unique i here
}

// p = exp(al - m[dst]) stored back into al; den += p
__global__ void k_expsum(const int* __restrict__ dst, float* __restrict__ al,
                         const float* __restrict__ m, float* __restrict__ den, int E, int H) {
  int e = blockIdx.x * blockDim.x + threadIdx.x;
  if (e >= E) return;
  int d = dst[e];
  for (int h = 0; h < H; ++h) {
    size_t i = (size_t)e * H + h;
    float p = __expf(al[i] - m[(size_t)d * H + h]);
    al[i] = p;
    atomicAdd(&den[(size_t)d * H + h], p);
  }
}

__global__ void k_expsum_self(float* __restrict__ alslf, const float* __restrict__ m,
                              float* __restrict__ den, int N, int H) {
  int i = blockIdx.x * blockDim.x + threadIdx.x;
  if (i >= N * H) return;
  float p = __expf(alslf[i] - m[i]);
  alslf[i] = p;
  den[i] += p;
}

// w = p / (den[dst] + eps)
__global__ void k_weights(const int* __restrict__ dst, float* __restrict__ al,
                          const float* __restrict__ den, int E, int H) {
  int e = blockIdx.x * blockDim.x + threadIdx.x;
  if (e >= E) return;
  int d = dst[e];
  for (int h = 0; h < H; ++h)
    al[(size_t)e * H + h] /= (den[(size_t)d * H + h] + 1e-16f);
}

__global__ void k_weights_self(float* __restrict__ alslf, const float* __restrict__ den,
                               int N, int H) {
  int i = blockIdx.x * blockDim.x + threadIdx.x;
  if (i >= N * H) return;
  alslf[i] /= (den[i] + 1e-16f);
}

// one block per edge, blockDim = H*C/2, float2 per thread: coalesced, no divides
__global__ void k_aggregate(const int* __restrict__ src, const int* __restrict__ dst,
                            const float* __restrict__ w, const float* __restrict__ hs,
                            float* __restrict__ out, int H, int C) {
  const int e  = blockIdx.x;
  const int j  = threadIdx.x * 2;
  const int HC = blockDim.x * 2;
  const int h  = j >= C ? 1 : 0;  // H == 2, C even
  const float ww = w[(size_t)e * H + h];
  const float2 hv = *(const float2*)&hs[(size_t)src[e] * HC + j];
  float* op = &out[(size_t)dst[e] * HC + j];
  atomicAdd(op + 0, ww * hv.x);
  atomicAdd(op + 1, ww * hv.y);
}

// one block per node; unique addresses; serialized after edge kernel on stream
__global__ void k_aggregate_self(const float* __restrict__ w, const float* __restrict__ hs,
                                 float* __restrict__ out, int H, int C) {
  const int n  = blockIdx.x;
  const int j  = threadIdx.x * 2;
  const int HC = blockDim.x * 2;
  const int h  = j >= C ? 1 : 0;
  const float ww = w[(size_t)n * H + h];
  const size_t i = (size_t)n * HC + j;
  const float2 hv = *(const float2*)&hs[i];
  float2 ov = *(const float2*)&out[i];
  ov.x += ww * hv.x;
  ov.y += ww * hv.y;
  *(float2*)&out[i] = ov;
}

__global__ void k_out_proj(const float* __restrict__ x, const float* __restrict__ w,
                           const float* __restrict__ b, float* __restrict__ out, int N, int K) {
  int n = blockIdx.x * blockDim.x + threadIdx.x;
  if (n >= N) return;
  const float* xr = x + (size_t)n * K;
  float acc = b[0];
  for (int k = 0; k < K; k += 4) {
    const float4 xv = *(const float4*)(xr + k);
    const float4 wv = *(const float4*)(w + k);
    acc += xv.x * wv.x + xv.y * wv.y + xv.z * wv.z + xv.w * wv.w;
  }
  out[n] = acc;
}

// ---------------- host orchestration ----------------
static inline unsigned cdivu(long a, long b) { return (unsigned)((a + b - 1) / b); }

static void gemm(const float* A, const float* B, const float* bias, float* C,
                 int M, int N, int K, float slope, int act, hipStream_t st) {
  dim3 g(cdivu(M, BM), cdivu(N, BN));
  k_gemm_wmma<<<g, TB, 0, st>>>(A, B, bias, C, M, N, K, slope, act);
}

struct ConvW {
  const float *Wsrc, *Wd, *We, *as_, *ad_, *ae_;
};

static void run_conv(const ConvW& cw,
                     const float* xs, int ns, int ks,
                     const float* xd, int nd, int kd,
                     const int* src, const int* dst, int E,
                     const float* ea, int ke,
                     int H, int C, bool self_loops,
                     float* out, float* hs, float* scr, hipStream_t st) {
  const int HC = H * C;
  float* vd    = scr; scr += H * kd;
  float* ve    = scr; scr += H * ke;
  float* ssrc  = scr; scr += (size_t)ns * H;
  float* sdst  = scr; scr += (size_t)nd * H;
  float* se    = scr; scr += (size_t)E * H;
  float* al    = scr; scr += (size_t)E * H;
  float* m     = scr; scr += (size_t)nd * H;
  float* den   = scr; scr += (size_t)nd * H;
  float* sesum = scr; scr += (size_t)nd * H;
  float* sslf  = scr; scr += (size_t)nd * H;
  float* alslf = scr; scr += (size_t)nd * H;
  float* cnt   = scr; scr += nd;

  // hs = x_src @ Wsrc  (f32 WMMA path)
  gemm(xs, cw.Wsrc, nullptr, hs, ns, HC, ks, 1.0f, 0, st);
  // collapsed attention vectors: vd = Wd_h @ a_d[h], ve = We_h @ a_e[h]
  k_make_v<<<cdivu(H * kd, TB), TB, 0, st>>>(cw.Wd, cw.ad_, vd, kd, C, H);
  k_make_v<<<cdivu(H * ke, TB), TB, 0, st>>>(cw.We, cw.ae_, ve, ke, C, H);
  k_scores_h2<<<cdivu(ns, TB), TB, 0, st>>>(hs, cw.as_, ssrc, ns, C);
  k_scores2<<<cdivu(nd, TB), TB, 0, st>>>(xd, vd, sdst, nd, kd);
  k_scores2<<<cdivu(E, TB), TB, 0, st>>>(ea, ve, se, E, ke);

  if (self_loops) {
    k_fill<<<cdivu((long)nd * H, TB), TB, 0, st>>>(sesum, 0.0f, (long)nd * H);
    k_fill<<<cdivu(nd, TB), TB, 0, st>>>(cnt, 0.0f, nd);
    k_selfloop_stats<<<cdivu(E, TB), TB, 0, st>>>(dst, se, sesum, cnt, E, H);
    k_selfloop_fin<<<cdivu((long)nd * H, TB), TB, 0, st>>>(sesum, cnt, sslf, nd, H);
  }

  k_fill<<<cdivu((long)nd * H, TB), TB, 0, st>>>(m, -3.0e38f, (long)nd * H);
  k_fill<<<cdivu((long)nd * H, TB), TB, 0, st>>>(den, 0.0f, (long)nd * H);

  k_logits_max<<<cdivu(E, TB), TB, 0, st>>>(src, dst, ssrc, sdst, se, al, m, E, H);
  if (self_loops)
    k_logits_max_self<<<cdivu((long)nd * H, TB), TB, 0, st>>>(ssrc, sdst, sslf, alslf, m, nd, H);

  k_expsum<<<cdivu(E, TB), TB, 0, st>>>(dst, al, m, den, E, H);
  if (self_loops)
    k_expsum_self<<<cdivu((long)nd * H, TB), TB, 0, st>>>(alslf, m, den, nd, H);

  k_weights<<<cdivu(E, TB), TB, 0, st>>>(dst, al, den, E, H);
  if (self_loops)
    k_weights_self<<<cdivu((long)nd * H, TB), TB, 0, st>>>(alslf, den, nd, H);

  k_aggregate<<<E, HC / 2, 0, st>>>(src, dst, al, hs, out, H, C);
  if (self_loops)
    k_aggregate_self<<<nd, HC / 2, 0, st>>>(alslf, hs, out, H, C);
}

extern "C" void kernel_launch(void* const* d_in, const int* in_sizes, int n_in,
                              void* d_out, int out_size, void* d_ws, size_t ws_size,
                              hipStream_t stream) {
  (void)n_in; (void)out_size; (void)ws_size;
  const int D_P = 34, D_B = 16, HID = 96, H = 2;

  const float* x_prop   = (const float*)d_in[0];
  const float* x_branch = (const float*)d_in[1];
  const int*   ei_pp    = (const int*)d_in[2];
  const int*   ei_bp    = (const int*)d_in[3];
  const int*   ei_bb    = (const int*)d_in[4];
  const float* ea_pp    = (const float*)d_in[5];
  const float* ea_bp    = (const float*)d_in[6];
  const float* ea_bb    = (const float*)d_in[7];

  const int NP  = in_sizes[0] / D_P;
  const int NB  = in_sizes[1] / D_B;
  const int Epp = in_sizes[2] / 2;
  const int Ebp = in_sizes[3] / 2;
  const int Ebb = in_sizes[4] / 2;

  // params: sorted-key pytree flatten
  const float* W_e_bb = (const float*)d_in[8];
  const float* W_e_bp = (const float*)d_in[9];
  const float* W_e_pp = (const float*)d_in[10];
  const float* W_in_b = (const float*)d_in[11];
  const float* W_in_p = (const float*)d_in[12];
  const float* W_out  = (const float*)d_in[13];
  const float* b_e_bb = (const float*)d_in[14];
  const float* b_e_bp = (const float*)d_in[15];
  const float* b_e_pp = (const float*)d_in[16];
  const float* b_in_b = (const float*)d_in[17];
  const float* b_in_p = (const float*)d_in[18];
  const float* b_out  = (const float*)d_in[19];
  const float* l1bb_W  = (const float*)d_in[20];
  const float* l1bb_We = (const float*)d_in[21];
  const float* l1bb_ad = (const float*)d_in[22];
  const float* l1bb_ae = (const float*)d_in[23];
  const float* l1bb_as = (const float*)d_in[24];
  const float* l1bb_b  = (const float*)d_in[25];
  const float* l1bp_Wd = (const float*)d_in[26];
  const float* l1bp_We = (const float*)d_in[27];
  const float* l1bp_Ws = (const float*)d_in[28];
  const float* l1bp_ad = (const float*)d_in[29];
  const float* l1bp_ae = (const float*)d_in[30];
  const float* l1bp_as = (const float*)d_in[31];
  const float* l1bp_b  = (const float*)d_in[32];
  const float* l1pp_W  = (const float*)d_in[33];
  const float* l1pp_We = (const float*)d_in[34];
  const float* l1pp_ad = (const float*)d_in[35];
  const float* l1pp_ae = (const float*)d_in[36];
  const float* l1pp_as = (const float*)d_in[37];
  const float* l1pp_b  = (const float*)d_in[38];
  const float* l2bb_W  = (const float*)d_in[39];
  const float* l2bb_We = (const float*)d_in[40];
  const float* l2bb_ad = (const float*)d_in[41];
  const float* l2bb_ae = (const float*)d_in[42];
  const float* l2bb_as = (const float*)d_in[43];
  const float* l2bb_b  = (const float*)d_in[44];
  const float* l2bp_Wd = (const float*)d_in[45];
  const float* l2bp_We = (const float*)d_in[46];
  const float* l2bp_Ws = (const float*)d_in[47];
  const float* l2bp_ad = (const float*)d_in[48];
  const float* l2bp_ae = (const float*)d_in[49];
  const float* l2bp_as = (const float*)d_in[50];
  const float* l2bp_b  = (const float*)d_in[51];
  const float* l2pp_W  = (const float*)d_in[52];
  const float* l2pp_We = (const float*)d_in[53];
  const float* l2pp_ad = (const float*)d_in[54];
  const float* l2pp_ae = (const float*)d_in[55];
  const float* l2pp_as = (const float*)d_in[56];
  const float* l2pp_b  = (const float*)d_in[57];

  // bump allocator over workspace
  char* wsb = (char*)d_ws;
  size_t off = 0;
  auto alloc = [&](size_t nfl) -> float* {
    float* p = (float*)(wsb + off);
    off += ((nfl * sizeof(float) + 255) / 256) * 256;
    return p;
  };
  float* xp0 = alloc((size_t)NP * HID);
  float* xb0 = alloc((size_t)NB * HID);
  float* epp = alloc((size_t)Epp * HID);
  float* ebp = alloc((size_t)Ebp * HID);
  float* ebb = alloc((size_t)Ebb * HID);
  float* xp1 = alloc((size_t)NP * 2 * HID);
  float* xb1 = alloc((size_t)NB * 2 * HID);
  float* xp2 = alloc((size_t)NP * 4 * HID);
  float* xb2 = alloc((size_t)NB * 4 * HID);
  float* hs  = alloc((size_t)NB * 4 * HID);
  float* scr = alloc((size_t)2 * 1000 * 1000);

  // ---- input + edge transforms (leaky_relu 0.01) ----
  gemm(x_prop,   W_in_p, b_in_p, xp0, NP,  HID, D_P, 0.01f, 1, stream);
  gemm(x_branch, W_in_b, b_in_b, xb0, NB,  HID, D_B, 0.01f, 1, stream);
  gemm(ea_pp,    W_e_pp, b_e_pp, epp, Epp, HID, 6,   0.01f, 1, stream);
  gemm(ea_bp,    W_e_bp, b_e_bp, ebp, Ebp, HID, 6,   0.01f, 1, stream);
  gemm(ea_bb,    W_e_bb, b_e_bb, ebb, Ebb, HID, 6,   0.01f, 1, stream);

  // ---- layer 1 (H=2, C=96, HC=192) ----
  {
    const int C = HID, HC = 2 * HID;
    k_init_bias2<<<cdivu((long)NP * HC, TB), TB, 0, stream>>>(xp1, l1pp_b, l1bp_b, NP, HC);
    k_init_bias2<<<cdivu((long)NB * HC, TB), TB, 0, stream>>>(xb1, l1bb_b, nullptr, NB, HC);
    ConvW pp{l1pp_W, l1pp_W, l1pp_We, l1pp_as, l1pp_ad, l1pp_ae};
    run_conv(pp, xp0, NP, HID, xp0, NP, HID, ei_pp, ei_pp + Epp, Epp, epp, HID,
             H, C, true, xp1, hs, scr, stream);
    ConvW bp{l1bp_Ws, l1bp_Wd, l1bp_We, l1bp_as, l1bp_ad, l1bp_ae};
    run_conv(bp, xb0, NB, HID, xp0, NP, HID, ei_bp, ei_bp + Ebp, Ebp, ebp, HID,
             H, C, false, xp1, hs, scr, stream);
    ConvW bb{l1bb_W, l1bb_W, l1bb_We, l1bb_as, l1bb_ad, l1bb_ae};
    run_conv(bb, xb0, NB, HID, xb0, NB, HID, ei_bb, ei_bb + Ebb, Ebb, ebb, HID,
             H, C, true, xb1, hs, scr, stream);
  }

  // ---- layer 2 (H=2, C=192, HC=384) ----
  {
    const int C = 2 * HID, HC = 4 * HID, DIN = 2 * HID;
    k_init_bias2<<<cdivu((long)NP * HC, TB), TB, 0, stream>>>(xp2, l2pp_b, l2bp_b, NP, HC);
    k_init_bias2<<<cdivu((long)NB * HC, TB), TB, 0, stream>>>(xb2, l2bb_b, nullptr, NB, HC);
    ConvW pp{l2pp_W, l2pp_W, l2pp_We, l2pp_as, l2pp_ad, l2pp_ae};
    run_conv(pp, xp1, NP, DIN, xp1, NP, DIN, ei_pp, ei_pp + Epp, Epp, epp, HID,
             H, C, true, xp2, hs, scr, stream);
    ConvW bp{l2bp_Ws, l2bp_Wd, l2bp_We, l2bp_as, l2bp_ad, l2bp_ae};
    run_conv(bp, xb1, NB, DIN, xp1, NP, DIN, ei_bp, ei_bp + Ebp, Ebp, ebp, HID,
             H, C, false, xp2, hs, scr, stream);
    ConvW bb{l2bb_W, l2bb_W, l2bb_We, l2bb_as, l2bb_ad, l2bb_ae};
    run_conv(bb, xb1, NB, DIN, xb1, NB, DIN, ei_bb, ei_bb + Ebb, Ebb, ebb, HID,
             H, C, true, xb2, hs, scr, stream);
  }

  // ---- output projection ----
  k_out_proj<<<cdivu(NP, TB), TB, 0, stream>>>(xp2, W_out, b_out, (float*)d_out, NP, 4 * HID);
}